// BispectrumCalculator_60292750901959
// MI455X (gfx1250) — compile-verified
//
#include <hip/hip_runtime.h>
#include <hip/hip_bf16.h>

// ---------------------------------------------------------------------------
// BispectrumCalculator for MI455X (gfx1250)
//   Stage 1: DFT of 32x1024 fp32 rows as a GEMM using V_WMMA_F32_16X16X4_F32
//            y_re = x * cos(2*pi*n*k/N), y_im = x * (-sin(...)), exact angle
//            reduction via t = (n*k) & 1023. Result -> d_ws as interleaved
//            float2 (re, im), 32*1024 complex = 256 KB.
//   Stage 2: streaming bispectrum write (256 MiB) with b128 non-temporal
//            stores; y row staged in LDS (8 KB) and reused for 4 rows j per block.
//   Stage 3: async d2d copy of `target` into the output tail.
// ---------------------------------------------------------------------------

typedef __attribute__((ext_vector_type(2))) float v2f;
typedef __attribute__((ext_vector_type(4))) float v4f;
typedef __attribute__((ext_vector_type(8))) float v8f;

#define NFFT 1024

// ---------------------------------------------------------------------------
// Stage 1: fp32 WMMA DFT.
// Grid: 32 blocks x 128 threads = 128 waves. Wave w covers output tile
// (m_tile = w & 1) x (n_tile = w >> 1): rows m_tile*16..+15 (batch dim),
// cols n_tile*16..+15 (frequency dim). K loop: 256 steps of K=4.
//
// VGPR layouts (CDNA5 ISA 7.12.2, f32 16x16x4):
//   A (16x4, 2 VGPRs): lanes 0-15 -> M=lane, K={0,1}; lanes 16-31 -> M=lane-16,
//                      K={2,3}; VGPR0 = first K of pair, VGPR1 = second.
//   B (4x16, 2 VGPRs): mirrored, N across lanes, lane half selects K pair.
//   C/D (16x16, 8 VGPRs): VGPR r -> M = r + 8*(lane>=16), N = lane & 15.
// ---------------------------------------------------------------------------
__global__ __launch_bounds__(128)
void dft_wmma_kernel(const float* __restrict__ x, float2* __restrict__ yws) {
  const int lane = threadIdx.x & 31;
  const int wid  = blockIdx.x * 4 + (threadIdx.x >> 5);   // 0..127
  const int m_tile = wid & 1;                             // batch-tile: 0..1
  const int n_tile = wid >> 1;                            // freq-tile:  0..63
  const int half = lane >> 4;                             // K-pair select
  const int l15  = lane & 15;

  const int arow = m_tile * 16 + l15;                     // A: batch row
  const int bcol = n_tile * 16 + l15;                     // B: frequency col
  const float wtheta = -6.283185307179586f / (float)NFFT;

  v8f acc_c = {0.f,0.f,0.f,0.f,0.f,0.f,0.f,0.f};          // accumulates Re(y)
  v8f acc_s = {0.f,0.f,0.f,0.f,0.f,0.f,0.f,0.f};          // accumulates Im(y)

  const float* __restrict__ xrow = x + arow * NFFT;

  for (int step = 0; step < NFFT / 4; ++step) {
    const int n0 = step * 4 + half * 2;                   // time index pair
    v2f a;
    a.x = xrow[n0];
    a.y = xrow[n0 + 1];

    // Exact modular angle reduction: exp(-2*pi*i*n*k/N) periodic in n*k mod N.
    const int t0 = (n0 * bcol) & (NFFT - 1);
    const int t1 = ((n0 + 1) * bcol) & (NFFT - 1);
    float c0, s0, c1, s1;
    __sincosf(wtheta * (float)t0, &s0, &c0);              // s = sin(-2pi t/N)
    __sincosf(wtheta * (float)t1, &s1, &c1);

    v2f bc; bc.x = c0; bc.y = c1;                         // cos   -> Re(y)
    v2f bs; bs.x = s0; bs.y = s1;                         // -sin  -> Im(y)

    acc_c = __builtin_amdgcn_wmma_f32_16x16x4_f32(
        false, a, false, bc, (short)0, acc_c, false, false);
    acc_s = __builtin_amdgcn_wmma_f32_16x16x4_f32(
        false, a, false, bs, (short)0, acc_s, false, false);
  }

  // D layout: VGPR r -> M = r + 8*half, N = l15.
  const int row_base = m_tile * 16 + 8 * half;
  const int col = n_tile * 16 + l15;
#pragma unroll
  for (int r = 0; r < 8; ++r) {
    float2 v;
    v.x = acc_c[r];
    v.y = acc_s[r];
    yws[(row_base + r) * NFFT + col] = v;
  }
}

// ---------------------------------------------------------------------------
// Stage 2: bispectrum planes. Store-bandwidth bound (256 MiB).
// Grid: 32 batches * 256 row-groups; block = 256 threads handles 4 rows j,
// each thread 4 consecutive k -> two 16B NT stores per row.
// ---------------------------------------------------------------------------
__global__ __launch_bounds__(256)
void bispec_kernel(const float2* __restrict__ Y, float* __restrict__ out) {
  __shared__ float2 yl[NFFT];

  const int b  = blockIdx.x >> 8;                 // 0..31
  const int j0 = (blockIdx.x & 255) << 2;         // row group of 4

  const float2* __restrict__ yb = Y + b * NFFT;
  for (int i = threadIdx.x; i < NFFT; i += 256) yl[i] = yb[i];
  __syncthreads();

  float* __restrict__ out_re = out + (size_t)b * 2 * NFFT * NFFT;
  float* __restrict__ out_im = out_re + (size_t)NFFT * NFFT;

  const int k0 = threadIdx.x << 2;

#pragma unroll
  for (int jj = 0; jj < 4; ++jj) {
    const int j = j0 + jj;
    const float2 A = yl[j];                       // y[j]
    v4f re, im;
#pragma unroll
    for (int e = 0; e < 4; ++e) {
      const int k = k0 + e;
      const float2 Yk = yl[k];                    // y[k]
      const float2 C  = yl[(k - j) & (NFFT - 1)]; // y[(k-j) mod N]
      // P = y[j] * conj(y[k])
      const float pr = A.x * Yk.x + A.y * Yk.y;
      const float pi = A.y * Yk.x - A.x * Yk.y;
      // Bx = P * C
      re[e] = pr * C.x - pi * C.y;
      im[e] = pr * C.y + pi * C.x;
    }
    __builtin_nontemporal_store(re, (v4f*)(out_re + j * NFFT + k0));
    __builtin_nontemporal_store(im, (v4f*)(out_im + j * NFFT + k0));
  }
}

// ---------------------------------------------------------------------------
extern "C" void kernel_launch(void* const* d_in, const int* in_sizes, int n_in,
                              void* d_out, int out_size, void* d_ws, size_t ws_size,
                              hipStream_t stream) {
  const float* x = (const float*)d_in[0];        // target: 32*1*1024 fp32
  float* out = (float*)d_out;                    // source planes + target tail
  float2* yws = (float2*)d_ws;                   // 32*1024 complex = 256 KB

  // Stage 1: DFT via fp32 WMMA (128 waves).
  dft_wmma_kernel<<<32, 128, 0, stream>>>(x, yws);

  // Stage 2: 256 MiB streaming bispectrum write.
  bispec_kernel<<<32 * 256, 256, 0, stream>>>(yws, out);

  // Stage 3: append target after the (32,2,1024,1024) source block.
  hipMemcpyAsync(out + (size_t)32 * 2 * NFFT * NFFT, x,
                 (size_t)32 * NFFT * sizeof(float),
                 hipMemcpyDeviceToDevice, stream);
}